// EfficientMemoryGELU_27006754358114
// MI455X (gfx1250) — compile-verified
//
#include <hip/hip_runtime.h>
#include <math.h>

// Exact-GELU (erf-based) streaming kernel for MI455X (gfx1250).
//
// Roofline: 2 * 4 B * 67.1M = 536.9 MB of HBM traffic -> ~23 us @ 23.3 TB/s.
// Round-1 disasm showed OCML erff is branchy (saveexec-divergent two-path
// polynomial); with N(0,1) inputs almost every wave executes BOTH paths,
// pushing VALU time above the memory floor. This version uses a branch-free
// Abramowitz-Stegun 7.1.26 erf (|abs err| <= 1.5e-7): 1 v_rcp_f32 +
// 1 v_exp_f32 + ~12 VALU ops per element, no divergence -> compute drops to
// ~8-10 us and the kernel is purely HBM-bandwidth-bound again.
//
// Data path: 128-bit non-temporal global loads/stores (stream-once data must
// not pollute WGP$/L2), 256-thread blocks (8 wave32 waves), one float4/thread.

typedef float v4f __attribute__((ext_vector_type(4)));

__device__ __forceinline__ float gelu_exact(float x) {
    // z = |x| / sqrt(2)
    const float z = fabsf(x) * 0.70710678118654752440f;

    // Abramowitz-Stegun 7.1.26: erf(z) = 1 - poly(t) * exp(-z^2),
    // t = 1/(1 + 0.3275911 z), max abs error 1.5e-7 for z >= 0.
    const float t = __builtin_amdgcn_rcpf(fmaf(0.3275911f, z, 1.0f));
    float p = fmaf(1.061405429f, t, -1.453152027f);
    p = fmaf(p, t, 1.421413741f);
    p = fmaf(p, t, -0.284496736f);
    p = fmaf(p, t, 0.254829592f);
    p = p * t;

    // exp(-z^2) = exp(-x^2/2) = exp2(-x^2 * log2(e)/2); arg <= ~26 so the raw
    // hardware v_exp_f32 needs no range reduction.
    const float e  = __builtin_amdgcn_exp2f(x * x * -0.72134752044448170368f);
    const float pe = p * e;                       // = 1 - erf(z) = erfc(z)

    // gelu(x) = 0.5 x (1 + erf(x/sqrt(2)))
    //         = 0.5 x (2 - pe)  for x >= 0
    //         = 0.5 x (pe)      for x <  0   (no 1+erf cancellation)
    const float g = (x >= 0.0f) ? (2.0f - pe) : pe;   // v_cndmask, branch-free
    return x * (0.5f * g);
}

__global__ __launch_bounds__(256, 1)
void gelu_f32_v4_kernel(const float* __restrict__ x,
                        float* __restrict__ y,
                        long long n) {
    const long long n4 = n >> 2;  // number of full float4 vectors
    const long long i  = (long long)blockIdx.x * (long long)blockDim.x + threadIdx.x;

    if (i < n4) {
        // 128-bit non-temporal load: global_load_b128 th:TH_LOAD_NT
        v4f v = __builtin_nontemporal_load(((const v4f*)x) + i);
        v4f r;
        r.x = gelu_exact(v.x);
        r.y = gelu_exact(v.y);
        r.z = gelu_exact(v.z);
        r.w = gelu_exact(v.w);
        // 128-bit non-temporal store: global_store_b128 th:TH_NT
        __builtin_nontemporal_store(r, ((v4f*)y) + i);
    }

    // Scalar tail for n not divisible by 4 (dead for 4*4096*4096, kept for
    // generality).
    const long long tail = n & 3LL;
    if (i < tail) {
        const long long t = (n4 << 2) + i;
        y[t] = gelu_exact(x[t]);
    }
}

extern "C" void kernel_launch(void* const* d_in, const int* in_sizes, int n_in,
                              void* d_out, int out_size, void* d_ws, size_t ws_size,
                              hipStream_t stream) {
    (void)n_in; (void)d_ws; (void)ws_size; (void)out_size;

    const float* x = (const float*)d_in[0];
    float*       y = (float*)d_out;
    const long long n = (long long)in_sizes[0];   // 4*4096*4096 = 67,108,864

    const int       threads = 256;                 // 8 wave32 waves per block
    const long long n4      = n >> 2;              // 16,777,216 float4s
    long long blocks = (n4 + threads - 1) / threads;  // 65,536 blocks
    if (blocks < 1) blocks = 1;

    gelu_f32_v4_kernel<<<dim3((unsigned)blocks), dim3(threads), 0, stream>>>(x, y, n);
}